// MultimodalHypFormer_66494683677181
// MI455X (gfx1250) — compile-verified
//
#include <hip/hip_runtime.h>

// Native vector type so __builtin_nontemporal_store lowers to
// global_store_b128 with th:TH_STORE_NT.
typedef float f32x4 __attribute__((ext_vector_type(4)));

// Problem constants (from the reference).
constexpr int B = 64;
constexpr int T = 512;
constexpr int I = 256;
constexpr int D = 768;
constexpr int S = T + I + 2;          // 770 tokens per sequence
constexpr int IMG_START_ID = 30522;   // TEXT_VOCAB - 2
constexpr int IMG_END_ID   = 30523;   // TEXT_VOCAB - 1
constexpr int D4 = D / 4;             // 192 float4 columns per row
constexpr int TOK_PER_BLOCK = 10;     // S = 77 * 10 -> no batch-crossing

// Grid: x = position tile (77), y = batch (64). One block = 192 threads
// (6 x wave32) streams 10 embedding rows (30 KB).
//
// Phase 1: load ALL token indices (clamped, always in-bounds) into SGPRs —
//          fully unrolled so the compiler can clause the s_loads and issue a
//          single s_wait_kmcnt 0, instead of a kmcnt drain per token.
// Phase 2: branchless (s_cselect) row/base resolution, one global_load_b128
//          gather per lane per token — all 10 gathers in flight — packed f32
//          add of the hoisted type row, non-temporal global_store_b128.
__global__ __launch_bounds__(D4) void embed_gather_kernel(
    const int*   __restrict__ text_ids,     // [B, T]
    const int*   __restrict__ image_tokens, // [B, I]
    const float* __restrict__ text_emb,     // [30524, D]
    const float* __restrict__ image_emb,    // [8192, D]
    const float* __restrict__ type_emb,     // [2, D]
    float*       __restrict__ out_emb)      // [B, S, D]
{
    const int c4   = (int)threadIdx.x;      // 0..191, float4 column
    const int b    = (int)blockIdx.y;       // batch index
    const int pos0 = (int)blockIdx.x * TOK_PER_BLOCK;

    // Hoist both type-embedding rows (6 KB total, L2/WGP$-resident).
    const f32x4 ty0 = *(const f32x4*)(type_emb + 4 * c4);
    const f32x4 ty1 = *(const f32x4*)(type_emb + D + 4 * c4);

    // ---- Phase 1: batch all index loads (uniform -> SMEM clause) ----
    int t_idx[TOK_PER_BLOCK];
    int i_idx[TOK_PER_BLOCK];
#pragma unroll
    for (int i = 0; i < TOK_PER_BLOCK; ++i) {
        const int pos  = pos0 + i;
        const int tpos = (pos < T) ? pos : (T - 1);          // clamp in-bounds
        int ipos = pos - (T + 1);
        ipos = (ipos < 0) ? 0 : ((ipos > I - 1) ? (I - 1) : ipos);
        t_idx[i] = text_ids[b * T + tpos];                   // s_load
        i_idx[i] = image_tokens[b * I + ipos];               // s_load
    }

    // ---- Phase 2: resolve rows, gather, add type, NT store ----
#pragma unroll
    for (int i = 0; i < TOK_PER_BLOCK; ++i) {
        const int pos    = pos0 + i;
        const bool is_img = (pos > T) & (pos <= T + I);

        const int row = (pos < T)  ? t_idx[i]
                      : (pos == T) ? IMG_START_ID
                      : is_img     ? i_idx[i]
                                   : IMG_END_ID;
        const float* base = is_img ? image_emb : text_emb;
        const float* src  = base + (size_t)row * D;

        f32x4 v = *(const f32x4*)(src + 4 * c4);   // global_load_b128 gather
        v = v + (is_img ? ty1 : ty0);              // add type embedding

        // Output is write-once: non-temporal store keeps the embedding
        // tables resident in the 192 MB L2 for the gather's reuse.
        const size_t token = (size_t)b * S + pos;
        __builtin_nontemporal_store(
            v, (f32x4*)(out_emb + token * D + 4 * c4));
    }
}

// Tail outputs: token_types (0/1) and attention_mask (all 1), as float.
// Grid: x = position tile, y = batch -> no modulo needed.
__global__ void embed_meta_kernel(float* __restrict__ out_types,
                                  float* __restrict__ out_mask)
{
    const int pos = (int)blockIdx.x * (int)blockDim.x + (int)threadIdx.x;
    if (pos >= S) return;
    const int idx = (int)blockIdx.y * S + pos;
    const float ty = (pos >= T + 1 && pos < T + 1 + I) ? 1.0f : 0.0f;
    __builtin_nontemporal_store(ty,   out_types + idx);
    __builtin_nontemporal_store(1.0f, out_mask  + idx);
}

extern "C" void kernel_launch(void* const* d_in, const int* in_sizes, int n_in,
                              void* d_out, int out_size, void* d_ws, size_t ws_size,
                              hipStream_t stream)
{
    (void)in_sizes; (void)n_in; (void)out_size; (void)d_ws; (void)ws_size;

    // Inputs in setup_inputs() order.
    const int*   text_ids     = (const int*)  d_in[0];
    const int*   image_tokens = (const int*)  d_in[1];
    const float* text_emb     = (const float*)d_in[2];
    const float* image_emb    = (const float*)d_in[3];
    const float* type_emb     = (const float*)d_in[4];

    // Outputs concatenated flat in return order (all as float).
    float* out       = (float*)d_out;
    float* out_emb   = out;                                  // B*S*D
    float* out_types = out + (size_t)B * S * D;              // B*S
    float* out_mask  = out_types + (size_t)B * S;            // B*S

    dim3 grid_gather(S / TOK_PER_BLOCK, B);                  // 77 x 64
    embed_gather_kernel<<<grid_gather, D4, 0, stream>>>(
        text_ids, image_tokens, text_emb, image_emb, type_emb, out_emb);

    dim3 grid_meta((S + 255) / 256, B);                      // 4 x 64
    embed_meta_kernel<<<grid_meta, 256, 0, stream>>>(out_types, out_mask);
}